// MFieldLayer_29411936043443
// MI455X (gfx1250) — compile-verified
//
#include <hip/hip_runtime.h>

// MI455X / gfx1250: wave32, V_WMMA_F32_16X16X4_F32 (full f32 precision, matches reference).
// One wave computes one cell:
//   out[t,cell,d] = (1/9) * sum_{h<9,k<8} x[t, nh_idx[cell,h], 8k+d] * basis[cell,h,k]
// basis k=0..7 = [0, 1, 0, 1, sin(.5dphi), cos(.5dphi), sin(.5dth), cos(.5dth)] * (1/9)
//
// WMMA mapping (matvec padded into 16x16x4, K-chunk = (h, k-half)):
//   A row 0 carries 4 basis values (rows 1..15 zero):
//     lane 0  -> (A[0][K0], A[0][K1]),  lane 16 -> (A[0][K2], A[0][K3])   [doc A 16x4 f32 layout]
//   B[c][N]  -> x_f32[t, idx(cell,h), (4*half + c)*8 + d], column N -> (t,d)
//     lane j: N = j&15, holds c = {0,1} (j<16) or {2,3} (j>=16)           [inferred, symmetric]
//   D row 0 (VGPR0, lanes 0..15) = out; two accumulators cover all 32 (t,d).
// 36 WMMAs/cell; kernel is HBM/L2-gather bound (~50MB unique x, L2-resident), WMMA is free margin.

typedef __attribute__((ext_vector_type(2))) float v2f;
typedef __attribute__((ext_vector_type(4))) float v4f;
typedef __attribute__((ext_vector_type(8))) float v8f;

#define NH 9
#define CPB 4   // cells per block: 128 threads = 4 waves

__global__ __launch_bounds__(128) void mfield_wmma_f32_kernel(
    const float* __restrict__ x,       // [4][n][64] f32
    const float* __restrict__ coords,  // [n][2]     f32
    const int*   __restrict__ nh_idx,  // [n][9]     i32
    float*       __restrict__ out,     // [4][n][8]  f32
    int n)
{
    __shared__ __align__(16) float xs[CPB][NH][4][64];  // gathered x rows, f32 (36864 B)
    __shared__ __align__(16) float bs[CPB][NH][8];      // basis * (1/9), f32 (1152 B)

    const int lane = threadIdx.x & 31;
    const int wave = threadIdx.x >> 5;
    const int cell = blockIdx.x * CPB + wave;           // grid sized exactly: n % CPB == 0
    const float inv = 1.0f / 9.0f;

    // ---------------- basis (lanes 0..8, one neighbor each) ----------------
    if (lane < NH) {
        const int   idxh = nh_idx[cell * NH + lane];
        const float dphi = coords[idxh * 2 + 0] - coords[cell * 2 + 0];
        const float dth  = coords[idxh * 2 + 1] - coords[cell * 2 + 1];
        float s1, c1, s2, c2;
        __sincosf(0.5f * dphi, &s1, &c1);
        __sincosf(0.5f * dth,  &s2, &c2);
        v4f lo, hi;
        lo[0] = 0.f;      lo[1] = inv;      lo[2] = 0.f;      lo[3] = inv;
        hi[0] = s1 * inv; hi[1] = c1 * inv; hi[2] = s2 * inv; hi[3] = c2 * inv;
        *(v4f*)&bs[wave][lane][0] = lo;
        *(v4f*)&bs[wave][lane][4] = hi;
    }

    // ---------------- gather x rows -> LDS (coalesced b128) ----------------
    // lane -> (t = lane/8, k = lane%8); each lane moves 8 contiguous f32.
    {
        const int t = lane >> 3;
        const int k = lane & 7;
        for (int h = 0; h < NH; ++h) {
            const int idxh = nh_idx[cell * NH + h];                 // wave-uniform -> s_load
            const float* src = x + ((size_t)t * n + idxh) * 64 + k * 8;
            *(v4f*)&xs[wave][h][t][k * 8 + 0] = *(const v4f*)(src);
            *(v4f*)&xs[wave][h][t][k * 8 + 4] = *(const v4f*)(src + 4);
        }
    }
    __syncthreads();

    // ---------------- fragment geometry ----------------
    const int  N        = lane & 15;           // B/D column
    const int  chalf    = (lane >> 4) & 1;     // B: lanes>=16 hold contraction c = {2,3}
    const bool rowlane  = (lane == 0) || (lane == 16);   // A row M=0 carriers
    const int  td0 = N, td1 = 16 + N;
    const int  t0 = td0 >> 3, d0 = td0 & 7;
    const int  t1 = td1 >> 3, d1 = td1 & 7;

    const float* xc = &xs[wave][0][0][0];
    const float* bc = &bs[wave][0][0];

    v8f acc0 = {};   // columns (t,d) = 0..15   -> t=0,1
    v8f acc1 = {};   // columns (t,d) = 16..31  -> t=2,3

    #pragma unroll
    for (int h = 0; h < NH; ++h) {
        #pragma unroll
        for (int half = 0; half < 2; ++half) {      // k = half*4 + c, c = 0..3
            // ---- A fragment (2 VGPRs): only row M=0 nonzero ----
            v2f a;
            if (half == 0) {
                // basis[k=0..3] = (0, inv, 0, inv): lanes 0 and 16 both hold (0, inv)
                a[0] = 0.f;
                a[1] = rowlane ? inv : 0.f;
            } else {
                // lane 0 -> basis[h][4..5], lane 16 -> basis[h][6..7] (broadcast ds_load_b64)
                const int kb = 4 + ((lane & 16) >> 3);
                const float b64a = bc[h * 8 + kb];
                const float b64b = bc[h * 8 + kb + 1];
                a[0] = rowlane ? b64a : 0.f;
                a[1] = rowlane ? b64b : 0.f;
            }

            // ---- B fragments (2 VGPRs each): shared base -> ds_load_2addr_b32 ----
            const int k0 = half * 4 + 2 * chalf;    // this lane's first k
            const float* r0 = xc + (h * 4 + t0) * 64;
            const float* r1 = xc + (h * 4 + t1) * 64;
            v2f b0, b1;
            b0[0] = r0[k0 * 8 + d0];
            b0[1] = r0[(k0 + 1) * 8 + d0];
            b1[0] = r1[k0 * 8 + d1];
            b1[1] = r1[(k0 + 1) * 8 + d1];

            // D = A*B + C  (v_wmma_f32_16x16x4_f32, f32 accumulate chain)
            acc0 = __builtin_amdgcn_wmma_f32_16x16x4_f32(false, a, false, b0,
                                                         (short)0, acc0, false, false);
            acc1 = __builtin_amdgcn_wmma_f32_16x16x4_f32(false, a, false, b1,
                                                         (short)0, acc1, false, false);
        }
    }

    // ---------------- store row M=0 of D (VGPR0 on lanes 0..15) ----------------
    if (lane < 16) {
        out[(size_t)t0 * n * 8 + (size_t)cell * 8 + d0] = acc0[0];
        out[(size_t)t1 * n * 8 + (size_t)cell * 8 + d1] = acc1[0];
    }
}

extern "C" void kernel_launch(void* const* d_in, const int* in_sizes, int n_in,
                              void* d_out, int out_size, void* d_ws, size_t ws_size,
                              hipStream_t stream) {
    const float* x      = (const float*)d_in[0];   // (1,1,4,n,64) f32
    const float* coords = (const float*)d_in[1];   // (n,2)        f32
    const int*   nh     = (const int*)  d_in[2];   // (n,9)        i32
    float*       out    = (float*)d_out;           // (1,1,4,n,8)  f32

    const int n = in_sizes[1] / 2;                 // coords is (n,2); n = 49152, n % 4 == 0
    const int blocks = n / CPB;                    // 12288 blocks x 128 threads (4 waves)
    mfield_wmma_f32_kernel<<<blocks, 128, 0, stream>>>(x, coords, nh, out, n);
}